// CoxPHLoss_33088428048880
// MI455X (gfx1250) — compile-verified
//
#include <hip/hip_runtime.h>
#include <hip/hip_bf16.h>
#include <stdint.h>

#define TMAXB 4096   // number of distinct time values
#define P2THREADS 512

static constexpr double FIX_SCALE = 4294967296.0;       // 2^32 fixed point
static constexpr double FIX_INV   = 1.0 / 4294967296.0;

typedef __attribute__((ext_vector_type(2))) float v2f;
typedef __attribute__((ext_vector_type(8))) float v8f;

#if defined(__has_builtin)
#  if __has_builtin(__builtin_amdgcn_wmma_f32_16x16x4_f32)
#    define HAVE_WMMA_F32_16X16X4 1
#  endif
#  if __has_builtin(__builtin_amdgcn_tensor_load_to_lds) && \
      __has_builtin(__builtin_amdgcn_s_wait_tensorcnt)
#    define HAVE_TDM 1
#  endif
#endif

#if defined(HAVE_TDM)
typedef __attribute__((ext_vector_type(4))) unsigned int u32x4;
typedef __attribute__((ext_vector_type(8))) int          i32x8;
typedef __attribute__((ext_vector_type(4))) int          i32x4;
#endif

// ---------------------------------------------------------------------------
// Workspace zero-init (harness does not re-poison between replays).
// Layout: u64 gExp[4096] | u64 gRiskFix[4096] | u32 gCnt[4096]  = 80 KiB
// ---------------------------------------------------------------------------
__global__ void coxph_zero_ws(unsigned int* __restrict__ ws32, int n32) {
  int i = blockIdx.x * blockDim.x + threadIdx.x;
  if (i < n32) ws32[i] = 0u;
}

// ---------------------------------------------------------------------------
// Phase 1: streaming histogram. LDS-privatized bins, ds atomics, then a
// fixed-point u64 global-atomic flush (deterministic cross-block combine).
// ---------------------------------------------------------------------------
__device__ __forceinline__ void coxph_acc1(float r, int t, int ev,
                                           float* sExp, float* sRisk,
                                           unsigned int* sCnt) {
  unsigned tt = ((unsigned)t) & (TMAXB - 1);
  float ex = __expf(r);                 // v_exp_f32 (TRANS, co-executes)
  atomicAdd(&sExp[tt], ex);             // ds_add_f32
  if (ev != 0) {
    atomicAdd(&sCnt[tt], 1u);           // ds_add_u32
    atomicAdd(&sRisk[tt], r);           // ds_add_f32
  }
}

__global__ __launch_bounds__(512) void coxph_hist(
    const float* __restrict__ risk,
    const int*   __restrict__ time,
    const int*   __restrict__ event,
    unsigned long long* __restrict__ gExp,
    unsigned long long* __restrict__ gRiskFix,
    unsigned int*       __restrict__ gCnt,
    int n) {
  __shared__ float        sExp[TMAXB];   // 16 KB
  __shared__ float        sRisk[TMAXB];  // 16 KB
  __shared__ unsigned int sCnt[TMAXB];   // 16 KB
  for (int i = threadIdx.x; i < TMAXB; i += blockDim.x) {
    sExp[i] = 0.0f; sRisk[i] = 0.0f; sCnt[i] = 0u;
  }
  __syncthreads();

  const int nv = n >> 2;
  const int stride = gridDim.x * blockDim.x;
  const float4* rp = (const float4*)risk;
  const int4*   tp = (const int4*)time;
  const int4*   ep = (const int4*)event;

  for (int v = blockIdx.x * blockDim.x + threadIdx.x; v < nv; v += stride) {
    float4 r = rp[v];                    // global_load_b128
    int4   t = tp[v];                    // global_load_b128
    int4   e = ep[v];                    // global_load_b128
    __builtin_prefetch(rp + v + stride, 0, 0);   // global_prefetch_b8
    __builtin_prefetch(tp + v + stride, 0, 0);
    __builtin_prefetch(ep + v + stride, 0, 0);
    coxph_acc1(r.x, t.x, e.x, sExp, sRisk, sCnt);
    coxph_acc1(r.y, t.y, e.y, sExp, sRisk, sCnt);
    coxph_acc1(r.z, t.z, e.z, sExp, sRisk, sCnt);
    coxph_acc1(r.w, t.w, e.w, sExp, sRisk, sCnt);
  }
  // scalar tail (N is a multiple of 4, but keep it general)
  for (int i = (nv << 2) + blockIdx.x * blockDim.x + threadIdx.x; i < n; i += stride)
    coxph_acc1(risk[i], time[i], event[i], sExp, sRisk, sCnt);

  __syncthreads();
  // Flush block-partials: convert to 2^32 fixed point, integer global atomics.
  for (int i = threadIdx.x; i < TMAXB; i += blockDim.x) {
    float se = sExp[i];
    if (se != 0.0f) {
      unsigned long long f = (unsigned long long)((double)se * FIX_SCALE);
      atomicAdd(&gExp[i], f);            // global_atomic_add_u64
    }
    unsigned c = sCnt[i];
    if (c != 0u) {
      atomicAdd(&gCnt[i], c);            // global_atomic_add_u32
      long long rf = (long long)((double)sRisk[i] * FIX_SCALE);
      atomicAdd(&gRiskFix[i], (unsigned long long)rf);  // two's-complement
    }
  }
}

// ---------------------------------------------------------------------------
// Phase 2: single workgroup. TDM stages the 32 KiB bucket array into LDS,
// suffix-sum over 4096 buckets (double), per-bucket D(t)*log(suffix) - E(t),
// then a block reduction whose last 16-wide step runs through
// V_WMMA_F32_16X16X4_F32 (all-ones A => D columns are sums over K).
// ---------------------------------------------------------------------------
__global__ __launch_bounds__(P2THREADS) void coxph_finish(
    const unsigned long long* __restrict__ gExp,
    const unsigned long long* __restrict__ gRiskFix,
    const unsigned int*       __restrict__ gCnt,
    float* __restrict__ out) {
  __shared__ double s[TMAXB];            // 32 KB; also reused as u64 staging
  constexpr int PER = TMAXB / P2THREADS; // 8
  const int tid = threadIdx.x;

  unsigned long long* raw = (unsigned long long*)s;
  unsigned long long tmp[PER];
#if defined(HAVE_TDM)
  if (tid < 32) {
    // Tensor DMA: one 4096x1 tile of 8-byte elements, global -> LDS.
    const unsigned long long ga = (unsigned long long)(uintptr_t)gExp;
    // Generic->LDS addrspace mapping keeps the byte offset in the low 32 bits.
    const unsigned int lds_off = (unsigned int)(uintptr_t)raw;
    u32x4 g0;
    g0.x = 1u;                                         // count=1 (valid D#)
    g0.y = lds_off;                                    // lds_addr (bytes)
    g0.z = (unsigned int)(ga & 0xFFFFFFFFu);           // global_addr[31:0]
    g0.w = (unsigned int)((ga >> 32) & 0x1FFFFFFu)     // global_addr[56:32]
           | (2u << 30);                               // type = 2 ("image")
    i32x8 g1;
    g1[0] = (int)(3u << 16);      // workgroup_mask=0, data_size=3 (8 bytes)
    g1[1] = (int)(4096u << 16);   // tensor_dim0[15:0] = 4096
    g1[2] = (int)(1u << 16);      // tensor_dim0[31:16]=0 | tensor_dim1 = 1
    g1[3] = (int)(4096u << 16);   // tensor_dim1 hi = 0  | tile_dim0 = 4096
    g1[4] = 1;                    // tile_dim1 = 1, tile_dim2 = 0
    g1[5] = 4096;                 // tensor_dim0_stride[31:0]
    g1[6] = 0;                    // stride hi bits / tensor_dim1_stride lo
    g1[7] = 0;                    // tensor_dim1_stride hi
    i32x4 gz4 = {0, 0, 0, 0};     // groups 2/3 unused (<=2D tensor)
    i32x8 gz8 = {0, 0, 0, 0, 0, 0, 0, 0};  // extra group (6-arg form), unused
    __builtin_amdgcn_tensor_load_to_lds(g0, g1, gz4, gz4, gz8, 0);
    __builtin_amdgcn_s_wait_tensorcnt(0);              // s_wait_tensorcnt 0
  }
  __syncthreads();
  for (int k = 0; k < PER; ++k) tmp[k] = raw[tid + k * P2THREADS];
  __syncthreads();
#else
  for (int k = 0; k < PER; ++k) tmp[k] = gExp[tid + k * P2THREADS];
#endif
  for (int k = 0; k < PER; ++k)
    s[tid + k * P2THREADS] = (double)tmp[k] * FIX_INV;   // back to real S(t)
  __syncthreads();

  // Reverse inclusive (suffix) scan, Hillis-Steele, 12 steps over 4096.
  for (int off = 1; off < TMAXB; off <<= 1) {
    double t2[PER];
    for (int k = 0; k < PER; ++k) {
      int i = tid + k * P2THREADS;
      double v = s[i];
      int j = i + off;
      if (j < TMAXB) v += s[j];
      t2[k] = v;
    }
    __syncthreads();
    for (int k = 0; k < PER; ++k) s[tid + k * P2THREADS] = t2[k];
    __syncthreads();
  }

  // Per-bucket contributions (double accumulation).
  double   local = 0.0;
  unsigned evt   = 0u;
  for (int k = 0; k < PER; ++k) {
    int i = tid + k * P2THREADS;
    unsigned c = gCnt[i];
    if (c != 0u) {               // c>0 => suffix>0, log is safe
      local += (double)c * log(s[i]);
      local -= (double)(long long)gRiskFix[i] * FIX_INV;
      evt   += c;
    }
  }

  // wave32 reduction (16 waves -> 16 partials)
  for (int sh = 16; sh > 0; sh >>= 1) {
    local += __shfl_down(local, sh);
    evt   += __shfl_down(evt, sh);
  }
  __shared__ float        wsum[16];
  __shared__ unsigned int wevt[16];
  const int lane = tid & 31, wv = tid >> 5;
  if (lane == 0) { wsum[wv] = (float)local; wevt[wv] = evt; }
  __syncthreads();

  if (wv == 0) {                 // full wave active: EXEC all-ones for WMMA
    unsigned ev = (lane < 16) ? wevt[lane] : 0u;
    for (int sh = 8; sh > 0; sh >>= 1) ev += __shfl_xor(ev, sh);

    float total;
#if defined(HAVE_WMMA_F32_16X16X4)
    // B[k][n]: partial w at (k=w&3, n=w>>2); A = ones(16x4)
    // => D[m][n] = sum_k B[k][n]; then 16-lane xor-reduce over n.
    const int n16 = lane & 15;
    const int k0  = (lane < 16) ? 0 : 2;   // B VGPR halves hold K={0,2}/{1,3}
    v2f aa = {1.0f, 1.0f};
    v2f bb = {0.0f, 0.0f};
    if (n16 < 4) { bb.x = wsum[k0 + 4 * n16]; bb.y = wsum[k0 + 1 + 4 * n16]; }
    v8f cc = {};
    v8f dd = __builtin_amdgcn_wmma_f32_16x16x4_f32(
        false, aa, false, bb, (short)0, cc, false, false);
    total = dd[0];
    for (int sh = 8; sh > 0; sh >>= 1) total += __shfl_xor(total, sh);
#else
    float p = (lane < 16) ? wsum[lane] : 0.0f;
    for (int sh = 8; sh > 0; sh >>= 1) p += __shfl_xor(p, sh);
    total = p;
#endif
    if (lane == 0) out[0] = (ev > 0u) ? total : 0.0f;
  }
}

// ---------------------------------------------------------------------------
extern "C" void kernel_launch(void* const* d_in, const int* in_sizes, int n_in,
                              void* d_out, int out_size, void* d_ws, size_t ws_size,
                              hipStream_t stream) {
  const float* risk  = (const float*)d_in[0];
  const int*   time  = (const int*)d_in[1];
  const int*   event = (const int*)d_in[2];
  const int n = in_sizes[0];

  unsigned long long* gExp     = (unsigned long long*)d_ws;
  unsigned long long* gRiskFix = gExp + TMAXB;
  unsigned int*       gCnt     = (unsigned int*)(gRiskFix + TMAXB);
  float* out = (float*)d_out;

  const int n32 = TMAXB * 2 /*u64*/ * 2 /*arrays*/ + TMAXB /*u32*/;  // dwords
  coxph_zero_ws<<<(n32 + 255) / 256, 256, 0, stream>>>((unsigned int*)d_ws, n32);

  // ~393K threads * 48 B/iter in flight => enough MLP for 23.3 TB/s stream.
  coxph_hist<<<768, 512, 0, stream>>>(risk, time, event, gExp, gRiskFix, gCnt, n);
  coxph_finish<<<1, P2THREADS, 0, stream>>>(gExp, gRiskFix, gCnt, out);
}